// LocalAttentionModule_V2_46359876993198
// MI455X (gfx1250) — compile-verified
//
#include <hip/hip_runtime.h>
#include <math.h>

// Problem constants (from reference): x = [8,128,512,512] f32
#define C_DIM 128
#define H_DIM 512
#define W_DIM 512
#define HW    (H_DIM * W_DIM)       // 262144 pixels / image
#define CHW   ((size_t)C_DIM * HW)
#define TS    128                   // tile side (512 / 4)
#define SEGS  16
#define LW    130                   // LDS row width: 128 + 1 zero col each side

typedef __attribute__((ext_vector_type(2))) float v2f;
typedef __attribute__((ext_vector_type(4))) float v4f;
typedef __attribute__((ext_vector_type(8))) float v8f;

// ---------------------------------------------------------------------------
// Pass 1 (HBM-bound, ~46us floor at 23.3 TB/s):
// l2[b,h,w] = sqrt(sum_c x[b,c,h,w]^2)
// One thread = 4 consecutive w positions (one B128 NT load per channel plane).
// x is streamed exactly once -> non-temporal so the 1 GiB stream doesn't
// flush L2; the 8 MB l2 output is stored temporally and pass 2 hits L2.
// ---------------------------------------------------------------------------
__global__ __launch_bounds__(256) void l2norm_stream_kernel(
    const float* __restrict__ x, float* __restrict__ l2) {
  size_t tid = (size_t)blockIdx.x * 256 + threadIdx.x;
  size_t img = tid >> 16;          // HW/4 == 65536 float4 per image
  size_t rem = tid & 65535;        // float4 index within image
  const float* base = x + img * CHW + rem * 4;

  float a0 = 0.f, a1 = 0.f, a2 = 0.f, a3 = 0.f;
#pragma unroll 8
  for (int c = 0; c < C_DIM; ++c) {
    v4f v = __builtin_nontemporal_load((const v4f*)(base + (size_t)c * HW));
    a0 = fmaf(v.x, v.x, a0);
    a1 = fmaf(v.y, v.y, a1);
    a2 = fmaf(v.z, v.z, a2);
    a3 = fmaf(v.w, v.w, a3);
  }
  v4f r;
  r.x = sqrtf(a0); r.y = sqrtf(a1); r.z = sqrtf(a2); r.w = sqrtf(a3);
  *(v4f*)(l2 + img * (size_t)HW + rem * 4) = r;
}

// ---------------------------------------------------------------------------
// Pass 2: per-tile 3x3 conv (zero pad) + bias + sigmoid via f32 WMMA.
// Block = (b, seg, 32-row quad). LDS tile is 34 x 130 with a pre-zeroed
// 1-row / 1-col halo, so every im2col tap is an unconditional ds_load
// (lo/hi K-slot selection is a cndmask on the LDS offset -> no exec
// divergence before the WMMAs, as the ISA requires).
// D(16x16) = A(16 px x 4 taps) x B(4 taps x 16, weights broadcast across
// columns) + C, accumulated over 3 K-chunks (taps 9..11 are zero).
// ---------------------------------------------------------------------------
__global__ __launch_bounds__(256) void conv_seg_wmma_kernel(
    const float* __restrict__ l2, const float* __restrict__ cw,
    const float* __restrict__ cb, float* __restrict__ out) {
  __shared__ float tile[34 * LW];

  int bid = blockIdx.x;
  int rq  = bid & 3;            // which 32-row quad of the tile
  int seg = (bid >> 2) & 15;
  int b   = bid >> 6;
  int r0  = rq * 32;
  int ti  = seg >> 2;           // tile coords in 4x4 grid
  int tj  = seg & 3;

  const float* img = l2 + (size_t)b * HW;

  // Stage 34 rows x 130 cols; halo rows/cols outside the tile are zero.
  for (int idx = threadIdx.x; idx < 34 * LW; idx += 256) {
    int rr = idx / LW;          // 0..33
    int cc = idx - rr * LW;     // 0..129
    int tr = r0 - 1 + rr;       // row within tile
    float v = 0.f;
    if (tr >= 0 && tr < TS && cc >= 1 && cc <= TS)
      v = img[(size_t)(ti * TS + tr) * W_DIM + tj * TS + (cc - 1)];
    tile[idx] = v;
  }

  // Weights (uniform per block -> scalar regs).
  float w0 = cw[seg * 9 + 0], w1 = cw[seg * 9 + 1], w2 = cw[seg * 9 + 2];
  float w3 = cw[seg * 9 + 3], w4 = cw[seg * 9 + 4], w5 = cw[seg * 9 + 5];
  float w6 = cw[seg * 9 + 6], w7 = cw[seg * 9 + 7], w8 = cw[seg * 9 + 8];
  float bias = cb[seg];

  __syncthreads();

  int  lane = threadIdx.x & 31;
  int  wave = threadIdx.x >> 5;
  int  px   = lane & 15;        // pixel (M) index this lane feeds
  bool hi   = lane >= 16;       // upper lane half -> K slots {2,3}

  // tap k lives at tile[base + (k/3)*LW + (k%3)] with base = rl*LW + c0+px
#define TOFF(k) (((k) / 3) * LW + ((k) % 3))

  // 32 rows x 8 col-groups = 256 WMMA groups per block, 32 per wave.
  for (int it = 0; it < 32; ++it) {
    int group = it * 8 + wave;
    int rl = group >> 3;              // local row 0..31
    int c0 = (group & 7) * 16;
    int base = rl * LW + c0 + px;
    int r = r0 + rl;

    v8f acc = {};

    // chunk 0: taps 0..3   (A layout: VGPR0 = K0|K2, VGPR1 = K1|K3)
    {
      v2f a, bm;
      a.x  = tile[base + (hi ? TOFF(2) : TOFF(0))];
      a.y  = tile[base + (hi ? TOFF(3) : TOFF(1))];
      bm.x = hi ? w2 : w0;
      bm.y = hi ? w3 : w1;
      acc = __builtin_amdgcn_wmma_f32_16x16x4_f32(false, a, false, bm,
                                                  (short)0, acc, false, false);
    }
    // chunk 1: taps 4..7
    {
      v2f a, bm;
      a.x  = tile[base + (hi ? TOFF(6) : TOFF(4))];
      a.y  = tile[base + (hi ? TOFF(7) : TOFF(5))];
      bm.x = hi ? w6 : w4;
      bm.y = hi ? w7 : w5;
      acc = __builtin_amdgcn_wmma_f32_16x16x4_f32(false, a, false, bm,
                                                  (short)0, acc, false, false);
    }
    // chunk 2: only tap 8 (taps 9..11 zero-padded)
    {
      float t8 = tile[base + TOFF(8)];   // unconditional load, then select
      v2f a, bm;
      a.x  = hi ? 0.f : t8;
      a.y  = 0.f;
      bm.x = hi ? 0.f : w8;
      bm.y = 0.f;
      acc = __builtin_amdgcn_wmma_f32_16x16x4_f32(false, a, false, bm,
                                                  (short)0, acc, false, false);
    }

    // All columns of D are identical. VGPR v holds M=v (lanes 0-15) and
    // M=v+8 (lanes 16-31): pixel p<8 written by lane p, pixel p>=8 by
    // lane 16+(p-8). Select acc[lane&7] via cndmask chain, one store.
    float val = acc[0];
#pragma unroll
    for (int v = 1; v < 8; ++v) val = ((lane & 7) == v) ? acc[v] : val;

    if ((lane & 15) < 8) {
      int pixel = (lane & 7) + (hi ? 8 : 0);
      float y = val + bias;
      float s = __builtin_amdgcn_rcpf(1.f + __expf(-y));  // v_rcp_f32
      size_t o = (size_t)b * HW + (size_t)seg * (TS * TS) +
                 (size_t)r * TS + c0 + pixel;
      __builtin_nontemporal_store(s, out + o);            // write-once output
    }
  }
#undef TOFF
}

// ---------------------------------------------------------------------------
// Harness entry. d_in: x [8,128,512,512] f32, conv_w [16,1,3,3] f32,
// conv_b [16] f32. d_out: [8,1,512,512] f32. d_ws: needs B*HW*4 = 8 MB.
// ---------------------------------------------------------------------------
extern "C" void kernel_launch(void* const* d_in, const int* in_sizes, int n_in,
                              void* d_out, int out_size, void* d_ws, size_t ws_size,
                              hipStream_t stream) {
  const float* x  = (const float*)d_in[0];
  const float* cw = (const float*)d_in[1];
  const float* cb = (const float*)d_in[2];
  float* out   = (float*)d_out;
  float* l2buf = (float*)d_ws;                 // B*HW floats = 8 MB scratch

  int B = (int)((size_t)in_sizes[0] / CHW);    // 8
  int blocks1 = (B * (HW / 4)) / 256;          // 2048
  int blocks2 = B * SEGS * 4;                  // 512 (4 row-quads per tile)

  l2norm_stream_kernel<<<blocks1, 256, 0, stream>>>(x, l2buf);
  conv_seg_wmma_kernel<<<blocks2, 256, 0, stream>>>(l2buf, cw, cb, out);
}